// GAT_61083024884000
// MI455X (gfx1250) — compile-verified
//
#include <hip/hip_runtime.h>
#include <math.h>

#define N_NODES 50000
#define E_EDGES 800000
#define ETOT    (E_EDGES + N_NODES)
#define DIN     128
#define DH1     16
#define H1      8
#define D1      (H1 * DH1)   // 128
#define DOUT    64

typedef float v2f __attribute__((ext_vector_type(2)));
typedef float v8f __attribute__((ext_vector_type(8)));

// ---------------------------------------------------------------- utilities

__global__ void gat_fill_kernel(float* __restrict__ p, int n, float v) {
    int i = blockIdx.x * blockDim.x + threadIdx.x;
    if (i < n) p[i] = v;
}

__device__ inline void atomicMaxF(float* addr, float val) {
#if defined(__HIP_DEVICE_COMPILE__) && __has_builtin(__hip_atomic_fetch_max)
    __hip_atomic_fetch_max(addr, val, __ATOMIC_RELAXED, __HIP_MEMORY_SCOPE_AGENT);
#else
    unsigned int* ua = (unsigned int*)addr;
    unsigned int old = *ua;
    while (__uint_as_float(old) < val) {
        unsigned int assumed = old;
        old = atomicCAS(ua, assumed, __float_as_uint(val));
        if (old == assumed) break;
    }
#endif
}

// ------------------------------------------------------- WMMA fp32 GEMM
// C[M x Nc] = A[M x K] * B[K x Nc], row-major, one wave32 per 16x16 tile.
// Uses V_WMMA_F32_16X16X4_F32 (fp32 in/out, no precision loss vs reference).
__global__ void gat_gemm_wmma_f32(const float* __restrict__ A,
                                  const float* __restrict__ B,
                                  float* __restrict__ C,
                                  int M, int Nc, int K) {
    const int colTiles   = Nc >> 4;
    const int totalTiles = (M >> 4) * colTiles;
    const int wave = blockIdx.x * (blockDim.x >> 5) + (threadIdx.x >> 5);
    if (wave >= totalTiles) return;            // wave-uniform: EXEC stays all-1s
    const int lane = threadIdx.x & 31;
    const int tr = wave / colTiles;
    const int tc = wave % colTiles;
    const int m  = lane & 15;                  // A-row / B-col within tile
    const int kh = (lane >> 4) << 1;           // K sub-offset: 0 or 2

    const float* arow  = A + (size_t)(tr * 16 + m) * K;
    const float* bbase = B + tc * 16 + m;

    v8f acc = {0.f, 0.f, 0.f, 0.f, 0.f, 0.f, 0.f, 0.f};
    for (int k = 0; k < K; k += 4) {
        v2f a = *(const v2f*)(arow + k + kh);          // A[m][k+kh], A[m][k+kh+1]
        v2f b;
        b.x = bbase[(size_t)(k + kh)     * Nc];        // B[k+kh][n]
        b.y = bbase[(size_t)(k + kh + 1) * Nc];        // B[k+kh+1][n]
#if defined(__HIP_DEVICE_COMPILE__)
#  if __has_builtin(__builtin_amdgcn_wmma_f32_16x16x4_f32)
        acc = __builtin_amdgcn_wmma_f32_16x16x4_f32(
            /*neg_a=*/false, a, /*neg_b=*/false, b,
            /*c_mod=*/(short)0, acc, /*reuse_a=*/false, /*reuse_b=*/false);
#  else
#    error "device pass: __builtin_amdgcn_wmma_f32_16x16x4_f32 not available"
#  endif
#else
        // host pass: never executed; keep semantics-free placeholder
        acc.s0 += a.x * b.x;
#endif
    }
    // D layout: VGPR j -> row j (lanes 0-15) / row j+8 (lanes 16-31), col = lane&15
    float* crow = C + (size_t)(tr * 16) * Nc + tc * 16 + m;
    const int rowoff = (lane >> 4) * 8;
#pragma unroll
    for (int j = 0; j < 8; ++j)
        crow[(size_t)(j + rowoff) * Nc] = acc[j];
}

// ---------------------------------------------- per-(node,head) attention dots
template <int HEADS, int DHD>
__global__ void gat_alpha_kernel(const float* __restrict__ h,
                                 const float* __restrict__ a_src,
                                 const float* __restrict__ a_dst,
                                 float* __restrict__ osrc,
                                 float* __restrict__ odst, int n) {
    int idx = blockIdx.x * blockDim.x + threadIdx.x;
    if (idx >= n * HEADS) return;
    int node = idx / HEADS;
    int hd   = idx % HEADS;
    const float* hv = h + (size_t)node * HEADS * DHD + hd * DHD;
    const float* as = a_src + hd * DHD;
    const float* ad = a_dst + hd * DHD;
    float ss = 0.f, sd = 0.f;
#pragma unroll
    for (int d = 0; d < DHD; ++d) {
        float v = hv[d];
        ss += v * as[d];
        sd += v * ad[d];
    }
    osrc[idx] = ss;
    odst[idx] = sd;
}

// ---------------------------------------------------------------- edge passes
__device__ inline void edge_nodes(int e, const int* __restrict__ src,
                                  const int* __restrict__ dst, int& s, int& d) {
    if (e < E_EDGES) { s = src[e]; d = dst[e]; }
    else             { s = e - E_EDGES; d = s; }   // self-loops appended
}

template <int HEADS>
__global__ void gat_edge_max_kernel(const int* __restrict__ src,
                                    const int* __restrict__ dst,
                                    const float* __restrict__ asrc,
                                    const float* __restrict__ adst,
                                    float* __restrict__ mdst) {
    int idx = blockIdx.x * blockDim.x + threadIdx.x;
    if (idx >= ETOT * HEADS) return;
    int e = idx / HEADS, hd = idx % HEADS;
    int s, d; edge_nodes(e, src, dst, s, d);
    float ev = asrc[s * HEADS + hd] + adst[d * HEADS + hd];
    ev = (ev > 0.f) ? ev : 0.2f * ev;                       // leaky_relu(0.2)
    atomicMaxF(&mdst[d * HEADS + hd], ev);
}

template <int HEADS>
__global__ void gat_edge_sum_kernel(const int* __restrict__ src,
                                    const int* __restrict__ dst,
                                    const float* __restrict__ asrc,
                                    const float* __restrict__ adst,
                                    const float* __restrict__ mdst,
                                    float* __restrict__ ssum) {
    int idx = blockIdx.x * blockDim.x + threadIdx.x;
    if (idx >= ETOT * HEADS) return;
    int e = idx / HEADS, hd = idx % HEADS;
    int s, d; edge_nodes(e, src, dst, s, d);
    float ev = asrc[s * HEADS + hd] + adst[d * HEADS + hd];
    ev = (ev > 0.f) ? ev : 0.2f * ev;
    float ex = __expf(ev - mdst[d * HEADS + hd]);
    atomicAdd(&ssum[d * HEADS + hd], ex);
}

// wave32 per edge: lanes cover the DIM feature dwords (coalesced atomics).
template <int DIM, int HEADS>
__global__ void gat_edge_agg_kernel(const int* __restrict__ src,
                                    const int* __restrict__ dst,
                                    const float* __restrict__ asrc,
                                    const float* __restrict__ adst,
                                    const float* __restrict__ mdst,
                                    const float* __restrict__ ssum,
                                    const float* __restrict__ h,
                                    float* __restrict__ agg) {
    int wave = blockIdx.x * (blockDim.x >> 5) + (threadIdx.x >> 5);
    if (wave >= ETOT) return;
    int lane = threadIdx.x & 31;
    int s, d; edge_nodes(wave, src, dst, s, d);
    __builtin_prefetch(&h[(size_t)s * DIM], 0, 1);   // global_prefetch_b8
    constexpr int PER = DIM / 32;
    constexpr int DHD = DIM / HEADS;
#pragma unroll
    for (int t = 0; t < PER; ++t) {
        int dim = lane + 32 * t;
        int hd  = dim / DHD;
        float ev = asrc[s * HEADS + hd] + adst[d * HEADS + hd];
        ev = (ev > 0.f) ? ev : 0.2f * ev;
        float alpha = __expf(ev - mdst[d * HEADS + hd]) /
                      (ssum[d * HEADS + hd] + 1e-16f);
        atomicAdd(&agg[(size_t)d * DIM + dim], h[(size_t)s * DIM + dim] * alpha);
    }
}

// ---------------------------------------------------------------- pointwise
__global__ void gat_elu_bias_kernel(float* __restrict__ x,
                                    const float* __restrict__ b, int total, int dim) {
    int i = blockIdx.x * blockDim.x + threadIdx.x;
    if (i >= total) return;
    float v = x[i] + b[i % dim];
    x[i] = (v > 0.f) ? v : (__expf(v) - 1.f);
}

// wave32 per node: bias + write h, then log_softmax over 64 classes.
__global__ void gat_final_kernel(const float* __restrict__ agg2,
                                 const float* __restrict__ b2,
                                 float* __restrict__ out_h,
                                 float* __restrict__ out_ls) {
    int wave = blockIdx.x * (blockDim.x >> 5) + (threadIdx.x >> 5);
    if (wave >= N_NODES) return;
    int lane = threadIdx.x & 31;
    const float* row = agg2 + (size_t)wave * DOUT;
    float v0 = row[lane]      + b2[lane];
    float v1 = row[lane + 32] + b2[lane + 32];
    out_h[(size_t)wave * DOUT + lane]      = v0;
    out_h[(size_t)wave * DOUT + lane + 32] = v1;
    float mx = fmaxf(v0, v1);
    for (int off = 16; off > 0; off >>= 1) mx = fmaxf(mx, __shfl_xor(mx, off, 32));
    float se = __expf(v0 - mx) + __expf(v1 - mx);
    for (int off = 16; off > 0; off >>= 1) se += __shfl_xor(se, off, 32);
    float lse = mx + __logf(se);
    out_ls[(size_t)wave * DOUT + lane]      = v0 - lse;
    out_ls[(size_t)wave * DOUT + lane + 32] = v1 - lse;
}

// ---------------------------------------------------------------- launcher
extern "C" void kernel_launch(void* const* d_in, const int* in_sizes, int n_in,
                              void* d_out, int out_size, void* d_ws, size_t ws_size,
                              hipStream_t stream) {
    const float* x      = (const float*)d_in[0];
    const int*   ei     = (const int*)d_in[1];
    const int*   src    = ei;
    const int*   dst    = ei + E_EDGES;
    const float* W1     = (const float*)d_in[2];
    const float* a_src1 = (const float*)d_in[3];
    const float* a_dst1 = (const float*)d_in[4];
    const float* b1     = (const float*)d_in[5];
    const float* W2     = (const float*)d_in[6];
    const float* a_src2 = (const float*)d_in[7];
    const float* a_dst2 = (const float*)d_in[8];
    const float* b2     = (const float*)d_in[9];

    float* ws   = (float*)d_ws;
    float* h1   = ws;  ws += (size_t)N_NODES * D1;
    float* agg1 = ws;  ws += (size_t)N_NODES * D1;
    float* h2   = ws;  ws += (size_t)N_NODES * DOUT;
    float* agg2 = ws;  ws += (size_t)N_NODES * DOUT;
    float* as1  = ws;  ws += (size_t)N_NODES * H1;
    float* ad1  = ws;  ws += (size_t)N_NODES * H1;
    float* m1   = ws;  ws += (size_t)N_NODES * H1;
    float* s1   = ws;  ws += (size_t)N_NODES * H1;
    float* as2  = ws;  ws += N_NODES;
    float* ad2  = ws;  ws += N_NODES;
    float* m2   = ws;  ws += N_NODES;
    float* s2   = ws;  ws += N_NODES;

    float* out_h  = (float*)d_out;
    float* out_ls = out_h + (size_t)N_NODES * DOUT;

    const int B = 256;
    const float NEGINF = -3.402823466e38f;
    #define GRID(n) ((int)(((long)(n) + B - 1) / B))

    // init accumulators
    gat_fill_kernel<<<GRID(N_NODES * H1), B, 0, stream>>>(m1, N_NODES * H1, NEGINF);
    gat_fill_kernel<<<GRID(N_NODES * H1), B, 0, stream>>>(s1, N_NODES * H1, 0.f);
    gat_fill_kernel<<<GRID(N_NODES * D1), B, 0, stream>>>(agg1, N_NODES * D1, 0.f);
    gat_fill_kernel<<<GRID(N_NODES), B, 0, stream>>>(m2, N_NODES, NEGINF);
    gat_fill_kernel<<<GRID(N_NODES), B, 0, stream>>>(s2, N_NODES, 0.f);
    gat_fill_kernel<<<GRID(N_NODES * DOUT), B, 0, stream>>>(agg2, N_NODES * DOUT, 0.f);

    // ---- layer 1
    int tiles1 = (N_NODES / 16) * (D1 / 16);
    gat_gemm_wmma_f32<<<(tiles1 + 7) / 8, 256, 0, stream>>>(x, W1, h1, N_NODES, D1, DIN);
    gat_alpha_kernel<H1, DH1><<<GRID(N_NODES * H1), B, 0, stream>>>(h1, a_src1, a_dst1,
                                                                    as1, ad1, N_NODES);
    gat_edge_max_kernel<H1><<<GRID((long)ETOT * H1), B, 0, stream>>>(src, dst, as1, ad1, m1);
    gat_edge_sum_kernel<H1><<<GRID((long)ETOT * H1), B, 0, stream>>>(src, dst, as1, ad1, m1, s1);
    gat_edge_agg_kernel<D1, H1><<<(ETOT + 7) / 8, 256, 0, stream>>>(src, dst, as1, ad1,
                                                                    m1, s1, h1, agg1);
    gat_elu_bias_kernel<<<GRID((long)N_NODES * D1), B, 0, stream>>>(agg1, b1,
                                                                    N_NODES * D1, D1);

    // ---- layer 2
    int tiles2 = (N_NODES / 16) * (DOUT / 16);
    gat_gemm_wmma_f32<<<(tiles2 + 7) / 8, 256, 0, stream>>>(agg1, W2, h2, N_NODES, DOUT, D1);
    gat_alpha_kernel<1, DOUT><<<GRID(N_NODES), B, 0, stream>>>(h2, a_src2, a_dst2,
                                                               as2, ad2, N_NODES);
    gat_edge_max_kernel<1><<<GRID(ETOT), B, 0, stream>>>(src, dst, as2, ad2, m2);
    gat_edge_sum_kernel<1><<<GRID(ETOT), B, 0, stream>>>(src, dst, as2, ad2, m2, s2);
    gat_edge_agg_kernel<DOUT, 1><<<(ETOT + 7) / 8, 256, 0, stream>>>(src, dst, as2, ad2,
                                                                     m2, s2, h2, agg2);
    gat_final_kernel<<<(N_NODES + 7) / 8, 256, 0, stream>>>(agg2, b2, out_h, out_ls);
    #undef GRID
}